// GIN_12317966204980
// MI455X (gfx1250) — compile-verified
//
#include <hip/hip_runtime.h>
#include <math.h>

// ---------------------------------------------------------------------------
// GIN forward (5x GINConv(eps=0) -> MLP(64->64->64) -> BN(train) -> ReLU),
// global mean pool, FC 64->64 ReLU, FC 64->10, log_softmax.
// fp32 WMMA (v_wmma_f32_16x16x4_f32) for all node-feature GEMMs.
// ---------------------------------------------------------------------------

#define NN      50000
#define NE      800000
#define DD      64
#define NG      512
#define NC      10
#define BN_EPS  1e-5f

typedef float v2f __attribute__((ext_vector_type(2)));
typedef float v8f __attribute__((ext_vector_type(8)));

// h = x ; agg = x   (agg pre-seeded with residual term for the scatter-add)
__global__ void init_h_agg(const float* __restrict__ x,
                           float* __restrict__ h, float* __restrict__ agg) {
    int i = blockIdx.x * blockDim.x + threadIdx.x;          // float4 index
    if (i < NN * DD / 4) {
        float4 v = ((const float4*)x)[i];
        ((float4*)h)[i]   = v;
        ((float4*)agg)[i] = v;
    }
}

__global__ void zero_f32(float* __restrict__ p, int n) {
    int i = blockIdx.x * blockDim.x + threadIdx.x;
    if (i < n) p[i] = 0.0f;
}

// agg[dst] += h[src]  : one thread per (edge, float4 chunk). L2-resident atomics.
__global__ void edge_scatter(const int* __restrict__ ei,
                             const float* __restrict__ h,
                             float* __restrict__ agg) {
    int t = blockIdx.x * blockDim.x + threadIdx.x;
    if (t >= NE * 16) return;
    int e = t >> 4, q = t & 15;
    int s = ei[e];
    int d = ei[NE + e];
    const float4 v = *(const float4*)(h + (size_t)s * DD + q * 4);
    float* dp = agg + (size_t)d * DD + q * 4;
    atomicAdd(dp + 0, v.x); atomicAdd(dp + 1, v.y);
    atomicAdd(dp + 2, v.z); atomicAdd(dp + 3, v.w);
}

// Fused: H2 = relu(X@Wa + ba) @ Wb + bb ; accumulate per-column sum / sumsq.
// 5 waves/block, each wave owns one 16-row strip (16x64 output = 4 WMMA tiles).
// 3125 row tiles == 625 blocks * 5 waves exactly -> no guards, EXEC always ~0.
__launch_bounds__(160)
__global__ void gin_mlp(const float* __restrict__ X,    // NN x 64 (agg)
                        const float* __restrict__ WaG,  // 64 x 64
                        const float* __restrict__ baG,  // 64
                        const float* __restrict__ WbG,  // 64 x 64
                        const float* __restrict__ bbG,  // 64
                        float* __restrict__ H2,         // NN x 64 (pre-BN out)
                        float* __restrict__ gstats) {   // [0:64)=sum, [64:128)=sumsq
    // Weights staged interleaved so a lane's (k, k+1) pair is one ds_load_b64:
    //   W2[(k>>1)*128 + n*2 + (k&1)] = W[k][n]
    __shared__ float Wa2[4096];
    __shared__ float Wb2[4096];
    __shared__ float baS[64], bbS[64];
    __shared__ float stage[5 * 16 * 64];   // per-wave C-layout -> A-layout tile
    __shared__ float csum[64], csq[64];

    const int tid = threadIdx.x;
    for (int idx = tid; idx < 4096; idx += 160) {
        int k = idx >> 6, n = idx & 63;
        int p = ((k >> 1) << 7) + (n << 1) + (k & 1);
        Wa2[p] = WaG[idx];
        Wb2[p] = WbG[idx];
    }
    if (tid < 64) { baS[tid] = baG[tid]; bbS[tid] = bbG[tid];
                    csum[tid] = 0.0f;    csq[tid] = 0.0f; }
    __syncthreads();

    const int wave = tid >> 5;
    const int lane = tid & 31;
    const int half = lane >> 4;      // ISA: lanes 0-15 hold K=0..1, 16-31 hold K=2..3
    const int lm   = lane & 15;
    const size_t m0 = ((size_t)blockIdx.x * 5 + wave) * 16;

    // ---------------- GEMM1: Z = relu(X @ Wa + ba) ----------------
    v8f acc[4] = {};
    const float* Xrow = X + (m0 + lm) * DD;
    #pragma unroll
    for (int kk = 0; kk < 16; ++kk) {
        const int kbase = kk * 4 + half * 2;
        v2f a = *(const v2f*)(Xrow + kbase);                       // global_load_b64
        const float* wrow = Wa2 + ((kbase >> 1) << 7) + (lm << 1); // ds_load_b64
        v2f b0 = *(const v2f*)(wrow +  0);
        v2f b1 = *(const v2f*)(wrow + 32);
        v2f b2 = *(const v2f*)(wrow + 64);
        v2f b3 = *(const v2f*)(wrow + 96);
        acc[0] = __builtin_amdgcn_wmma_f32_16x16x4_f32(false, a, false, b0, (short)0, acc[0], false, false);
        acc[1] = __builtin_amdgcn_wmma_f32_16x16x4_f32(false, a, false, b1, (short)0, acc[1], false, false);
        acc[2] = __builtin_amdgcn_wmma_f32_16x16x4_f32(false, a, false, b2, (short)0, acc[2], false, false);
        acc[3] = __builtin_amdgcn_wmma_f32_16x16x4_f32(false, a, false, b3, (short)0, acc[3], false, false);
    }

    // bias + relu, transpose C-layout tile into A-layout via per-wave LDS tile
    float* st = stage + wave * (16 * 64);
    #pragma unroll
    for (int nt = 0; nt < 4; ++nt) {
        const int c = nt * 16 + lm;
        const float bav = baS[c];
        #pragma unroll
        for (int r = 0; r < 8; ++r) {
            float z = fmaxf(acc[nt][r] + bav, 0.0f);
            st[(r + half * 8) * 64 + c] = z;     // row = vgpr + 8*half, col = lane%16
        }
    }
    __syncthreads();

    // ---------------- GEMM2: H2 = Z @ Wb + bb ----------------
    #pragma unroll
    for (int nt = 0; nt < 4; ++nt)
        #pragma unroll
        for (int r = 0; r < 8; ++r) acc[nt][r] = 0.0f;

    const float* Zrow = st + lm * DD;
    #pragma unroll
    for (int kk = 0; kk < 16; ++kk) {
        const int kbase = kk * 4 + half * 2;
        v2f a = *(const v2f*)(Zrow + kbase);                       // ds_load_b64
        const float* wrow = Wb2 + ((kbase >> 1) << 7) + (lm << 1);
        v2f b0 = *(const v2f*)(wrow +  0);
        v2f b1 = *(const v2f*)(wrow + 32);
        v2f b2 = *(const v2f*)(wrow + 64);
        v2f b3 = *(const v2f*)(wrow + 96);
        acc[0] = __builtin_amdgcn_wmma_f32_16x16x4_f32(false, a, false, b0, (short)0, acc[0], false, false);
        acc[1] = __builtin_amdgcn_wmma_f32_16x16x4_f32(false, a, false, b1, (short)0, acc[1], false, false);
        acc[2] = __builtin_amdgcn_wmma_f32_16x16x4_f32(false, a, false, b2, (short)0, acc[2], false, false);
        acc[3] = __builtin_amdgcn_wmma_f32_16x16x4_f32(false, a, false, b3, (short)0, acc[3], false, false);
    }

    // bias, write H2, reduce BN stats (block LDS -> 128 global atomics/block)
    #pragma unroll
    for (int nt = 0; nt < 4; ++nt) {
        const int c = nt * 16 + lm;
        const float bbv = bbS[c];
        float s = 0.0f, ss = 0.0f;
        #pragma unroll
        for (int r = 0; r < 8; ++r) {
            float v = acc[nt][r] + bbv;
            H2[(m0 + r + half * 8) * DD + c] = v;
            s += v; ss += v * v;
        }
        atomicAdd(&csum[c], s);
        atomicAdd(&csq[c], ss);
    }
    __syncthreads();
    if (tid < 64) {
        atomicAdd(&gstats[tid],      csum[tid]);
        atomicAdd(&gstats[64 + tid], csq[tid]);
    }
}

// BN(train stats) + ReLU, in place on h; also seeds agg for next layer's residual.
__global__ void bn_relu(const float* __restrict__ gstats,
                        const float* __restrict__ gamma,
                        const float* __restrict__ beta,
                        float* __restrict__ h, float* __restrict__ agg) {
    __shared__ float sc[64], sh[64];
    const int tid = threadIdx.x;
    if (tid < 64) {
        float mu  = gstats[tid] * (1.0f / NN);
        float var = gstats[64 + tid] * (1.0f / NN) - mu * mu;
        float s   = gamma[tid] * rsqrtf(var + BN_EPS);
        sc[tid] = s;
        sh[tid] = beta[tid] - mu * s;
    }
    __syncthreads();
    int i = blockIdx.x * blockDim.x + tid;                  // float4 index
    if (i < NN * DD / 4) {
        int c = (i & 15) * 4;                               // (i*4) % 64
        float4 v = ((const float4*)h)[i];
        v.x = fmaxf(v.x * sc[c + 0] + sh[c + 0], 0.0f);
        v.y = fmaxf(v.y * sc[c + 1] + sh[c + 1], 0.0f);
        v.z = fmaxf(v.z * sc[c + 2] + sh[c + 2], 0.0f);
        v.w = fmaxf(v.w * sc[c + 3] + sh[c + 3], 0.0f);
        ((float4*)h)[i]   = v;
        ((float4*)agg)[i] = v;
    }
}

// pooled[batch[i]] += h[i] ; cnt[batch[i]] += 1
__global__ void pool_scatter(const int* __restrict__ batch,
                             const float* __restrict__ h,
                             float* __restrict__ pooled, float* __restrict__ cnt) {
    int t = blockIdx.x * blockDim.x + threadIdx.x;
    if (t >= NN * 16) return;
    int i = t >> 4, q = t & 15;
    int g = batch[i];
    const float4 v = *(const float4*)(h + (size_t)i * DD + q * 4);
    float* dp = pooled + (size_t)g * DD + q * 4;
    atomicAdd(dp + 0, v.x); atomicAdd(dp + 1, v.y);
    atomicAdd(dp + 2, v.z); atomicAdd(dp + 3, v.w);
    if (q == 0) atomicAdd(cnt + g, 1.0f);
}

// per-graph head: mean, fc1+relu, fc2, log_softmax. One block (64 thr) per graph.
__global__ void head(const float* __restrict__ pooled, const float* __restrict__ cnt,
                     const float* __restrict__ fc1w, const float* __restrict__ fc1b,
                     const float* __restrict__ fc2w, const float* __restrict__ fc2b,
                     float* __restrict__ out) {
    __shared__ float row[64], o1[64], lg[NC], lse;
    const int g = blockIdx.x, t = threadIdx.x;
    float c = fmaxf(cnt[g], 1.0f);
    row[t] = pooled[(size_t)g * DD + t] / c;
    __syncthreads();
    float s = fc1b[t];
    #pragma unroll
    for (int k = 0; k < 64; ++k) s += row[k] * fc1w[k * 64 + t];
    o1[t] = fmaxf(s, 0.0f);
    __syncthreads();
    if (t < NC) {
        float l = fc2b[t];
        #pragma unroll
        for (int k = 0; k < 64; ++k) l += o1[k] * fc2w[k * NC + t];
        lg[t] = l;
    }
    __syncthreads();
    if (t == 0) {
        float m = -INFINITY;
        for (int j = 0; j < NC; ++j) m = fmaxf(m, lg[j]);
        float sum = 0.0f;
        for (int j = 0; j < NC; ++j) sum += expf(lg[j] - m);
        lse = m + logf(sum);
    }
    __syncthreads();
    if (t < NC) out[(size_t)g * NC + t] = lg[t] - lse;
}

extern "C" void kernel_launch(void* const* d_in, const int* in_sizes, int n_in,
                              void* d_out, int out_size, void* d_ws, size_t ws_size,
                              hipStream_t stream) {
    const float* x     = (const float*)d_in[0];
    const int*   ei    = (const int*)  d_in[1];
    const int*   batch = (const int*)  d_in[2];
    const float* Wa    = (const float*)d_in[3];
    const float* ba    = (const float*)d_in[4];
    const float* Wb    = (const float*)d_in[5];
    const float* bb    = (const float*)d_in[6];
    const float* gamma = (const float*)d_in[7];
    const float* beta  = (const float*)d_in[8];
    const float* fc1w  = (const float*)d_in[9];
    const float* fc1b  = (const float*)d_in[10];
    const float* fc2w  = (const float*)d_in[11];
    const float* fc2b  = (const float*)d_in[12];
    float* out = (float*)d_out;

    float* ws     = (float*)d_ws;
    float* h      = ws;                          // NN*DD
    float* agg    = h   + (size_t)NN * DD;       // NN*DD
    float* gstats = agg + (size_t)NN * DD;       // 128
    float* pooled = gstats + 128;                // NG*DD
    float* cnt    = pooled + (size_t)NG * DD;    // NG

    init_h_agg<<<(NN * DD / 4 + 255) / 256, 256, 0, stream>>>(x, h, agg);

    for (int l = 0; l < 5; ++l) {
        zero_f32<<<1, 128, 0, stream>>>(gstats, 128);
        edge_scatter<<<(NE * 16 + 255) / 256, 256, 0, stream>>>(ei, h, agg);
        gin_mlp<<<625, 160, 0, stream>>>(agg,
                                         Wa + (size_t)l * DD * DD, ba + (size_t)l * DD,
                                         Wb + (size_t)l * DD * DD, bb + (size_t)l * DD,
                                         h, gstats);
        bn_relu<<<(NN * DD / 4 + 255) / 256, 256, 0, stream>>>(
            gstats, gamma + (size_t)l * DD, beta + (size_t)l * DD, h, agg);
    }

    zero_f32<<<(NG * DD + NG + 255) / 256, 256, 0, stream>>>(pooled, NG * DD + NG);
    pool_scatter<<<(NN * 16 + 255) / 256, 256, 0, stream>>>(batch, h, pooled, cnt);
    head<<<NG, 64, 0, stream>>>(pooled, cnt, fc1w, fc1b, fc2w, fc2b, out);
}